// OrthogonalBayesianDense_10024453669626
// MI455X (gfx1250) — compile-verified
//
#include <hip/hip_runtime.h>
#include <hip/hip_bf16.h>

#define IN_F   4096
#define OUT_F  4096
#define ORDER  8
#define N_ROWS 8192

#define KC       32            // K columns staged per chunk
#define KSTRIDE  36            // padded LDS row stride (floats) -> conflict-free
#define WAVES    4             // waves per block in pass1

typedef float v2f __attribute__((ext_vector_type(2)));
typedef float v8f __attribute__((ext_vector_type(8)));

// ---------------------------------------------------------------------------
// Kernel 0: reparameterize v/d/bias, normalize v rows (pad to 16 rows with 0),
// build Gram matrix G of normalized rows, and M8 = 2*(I+S)^-1 with
// S[i][j] = 2*G[j][i] (j<i). Single block.
// ---------------------------------------------------------------------------
__global__ void obd_setup(const float* __restrict__ v_mean,
                          const float* __restrict__ v_logvar,
                          const float* __restrict__ d_mean,
                          const float* __restrict__ d_logvar,
                          const float* __restrict__ b_mean,
                          const float* __restrict__ b_logvar,
                          const float* __restrict__ eps_v,
                          const float* __restrict__ eps_d,
                          const float* __restrict__ eps_b,
                          float* __restrict__ vn,      // [16][IN_F] (rows 8..15 zero)
                          float* __restrict__ dvec,    // [IN_F]
                          float* __restrict__ bvec,    // [OUT_F]
                          float* __restrict__ M8) {    // [8][8]
    __shared__ float red[256];
    __shared__ float G[ORDER][ORDER];
    const int t = threadIdx.x;

    // --- reparameterized, normalized Householder vectors ---
    for (int i = 0; i < ORDER; ++i) {
        float ss = 0.0f;
        for (int c = t; c < IN_F; c += 256) {
            size_t idx = (size_t)i * IN_F + c;
            float val = v_mean[idx] + __expf(0.5f * v_logvar[idx]) * eps_v[idx];
            vn[idx] = val;
            ss += val * val;
        }
        red[t] = ss;
        __syncthreads();
        for (int s = 128; s > 0; s >>= 1) {
            if (t < s) red[t] += red[t + s];
            __syncthreads();
        }
        float inv = rsqrtf(red[0]);
        __syncthreads();
        for (int c = t; c < IN_F; c += 256)
            vn[(size_t)i * IN_F + c] *= inv;
        __syncthreads();
    }
    // zero-pad rows 8..15 so the WMMA B operand needs no lane masking
    for (int c = t; c < 8 * IN_F; c += 256)
        vn[(size_t)8 * IN_F + c] = 0.0f;

    // --- diag and bias samples ---
    for (int c = t; c < IN_F; c += 256)
        dvec[c] = d_mean[c] + __expf(0.5f * d_logvar[c]) * eps_d[c];
    for (int c = t; c < OUT_F; c += 256)
        bvec[c] = b_mean[c] + __expf(0.5f * b_logvar[c]) * eps_b[c];
    __syncthreads();

    // --- Gram matrix of normalized rows ---
    for (int j = 0; j < ORDER; ++j) {
        for (int i = j; i < ORDER; ++i) {
            float ss = 0.0f;
            for (int c = t; c < IN_F; c += 256)
                ss += vn[(size_t)j * IN_F + c] * vn[(size_t)i * IN_F + c];
            red[t] = ss;
            __syncthreads();
            for (int s = 128; s > 0; s >>= 1) {
                if (t < s) red[t] += red[t + s];
                __syncthreads();
            }
            if (t == 0) { G[j][i] = red[0]; G[i][j] = red[0]; }
            __syncthreads();
        }
    }

    // --- M8 = 2 * (I + S)^-1, forward substitution (unit lower triangular) ---
    if (t == 0) {
        float A[ORDER][ORDER];
        for (int i = 0; i < ORDER; ++i)
            for (int j = 0; j < ORDER; ++j)
                A[i][j] = (i == j) ? 1.0f : 0.0f;
        for (int i = 1; i < ORDER; ++i)
            for (int j = 0; j < i; ++j) {
                float s = 0.0f;
                for (int k = j; k < i; ++k)
                    s += 2.0f * G[k][i] * A[k][j];   // S[i][k] = 2*G[k][i]
                A[i][j] = -s;
            }
        for (int i = 0; i < ORDER; ++i)
            for (int j = 0; j < ORDER; ++j)
                M8[i * ORDER + j] = 2.0f * A[i][j];
    }
}

// ---------------------------------------------------------------------------
// Kernel 1: P = x @ Vn^T  (tall-skinny GEMM, N=8 padded to 16) via
// V_WMMA_F32_16X16X4_F32, with the x tile double-buffered into LDS using
// GLOBAL_LOAD_ASYNC_TO_LDS_B128 (ASYNCcnt-tracked, coalesced 512B/issue).
// Each wave owns one 16-row tile and its private LDS region -> no barriers,
// only s_wait_asynccnt. LDS rows padded to 36 floats => conflict-free
// ds_load_b64 A-fragment reads (bank = (36*m + k) mod 64, all distinct).
// ---------------------------------------------------------------------------
__global__ void obd_pass1(const float* __restrict__ x,
                          const float* __restrict__ vn,   // [16][IN_F], rows 8..15 zero
                          float* __restrict__ P) {        // [N_ROWS][8]
    __shared__ __align__(16) float xs[WAVES][2][16 * KSTRIDE];

    const int wave  = threadIdx.x >> 5;
    const int lane  = threadIdx.x & 31;
    const int tile  = blockIdx.x * WAVES + wave;   // 16-row tile index
    const int row0  = tile * 16;
    const int m     = lane & 15;                   // A: row in tile, B: column
    const int khalf = (lane >> 4) << 1;            // 0 or 2

    // Staging map: one b128 op = 32 lanes * 16B = 4 rows x 32 cols
    const int ld_row = lane >> 3;                  // 0..3
    const int ld_col = (lane & 7) << 2;            // 0,4,...,28

    const float* __restrict__ vrow = vn + (size_t)m * IN_F + khalf;

    unsigned lds_base0 = (unsigned)(unsigned long long)(uintptr_t)&xs[wave][0][0];
    unsigned lds_base1 = (unsigned)(unsigned long long)(uintptr_t)&xs[wave][1][0];

    // Issue 4 async b128 loads covering a 16 x KC chunk of this wave's tile.
    auto issue_chunk = [&](int k0, unsigned lds_base) {
#pragma unroll
        for (int g = 0; g < 4; ++g) {
            const int r = g * 4 + ld_row;
            unsigned lds_off = lds_base + (unsigned)((r * KSTRIDE + ld_col) * 4);
            unsigned long long gaddr = (unsigned long long)(uintptr_t)
                (x + (size_t)(row0 + r) * IN_F + k0 + ld_col);
            asm volatile("global_load_async_to_lds_b128 %0, %1, off"
                         :: "v"(lds_off), "v"(gaddr) : "memory");
        }
    };

    issue_chunk(0, lds_base0);
    issue_chunk(KC, lds_base1);

    v8f acc = {};
    for (int k0 = 0; k0 < IN_F; k0 += KC) {
        const int buf = (k0 / KC) & 1;
        // Async loads complete in order: <=4 outstanding means the current
        // buffer's 4 ops are done; the next buffer's 4 may still be in flight.
        asm volatile("s_wait_asynccnt 0x4" ::: "memory");

        const float* __restrict__ xsb = &xs[wave][buf][0];
#pragma unroll
        for (int kk = 0; kk < KC; kk += 4) {
            v2f a = *(const v2f*)(xsb + m * KSTRIDE + kk + khalf);
            v2f b = *(const v2f*)(vrow + k0 + kk);
            acc = __builtin_amdgcn_wmma_f32_16x16x4_f32(
                false, a, false, b, (short)0, acc, false, false);
        }

        const int knext = k0 + 2 * KC;
        if (knext < IN_F)
            issue_chunk(knext, buf ? lds_base1 : lds_base0);
    }
    asm volatile("s_wait_asynccnt 0x0" ::: "memory");

    // C/D layout: VGPR r holds M = r + 8*(lane>=16), N = lane&15.
    if (m < ORDER) {
        const int rbase = row0 + ((lane >> 4) << 3);
#pragma unroll
        for (int r = 0; r < 8; ++r)
            P[(size_t)(rbase + r) * ORDER + m] = acc[r];
    }
}

// ---------------------------------------------------------------------------
// Kernel 2: per row r: c = M8 * P[r];  out = (x - sum_i c_i*vn_i) * d + bias.
// One block per row; fully coalesced streaming pass. vn (128 KB) is L2-hot,
// and x is largely L2-resident from pass 1 (134 MB vs 192 MB L2).
// ---------------------------------------------------------------------------
__global__ void obd_pass2(const float* __restrict__ x,
                          const float* __restrict__ vn,
                          const float* __restrict__ P,
                          const float* __restrict__ M8,
                          const float* __restrict__ dvec,
                          const float* __restrict__ bvec,
                          float* __restrict__ out) {
    const int r = blockIdx.x;

    float p[ORDER], c[ORDER];
#pragma unroll
    for (int j = 0; j < ORDER; ++j) p[j] = P[(size_t)r * ORDER + j];
#pragma unroll
    for (int i = 0; i < ORDER; ++i) {
        float s = 0.0f;
#pragma unroll
        for (int j = 0; j < ORDER; ++j) s += M8[i * ORDER + j] * p[j];
        c[i] = s;
    }

    const float* __restrict__ xr = x + (size_t)r * IN_F;
    float* __restrict__ outr = out + (size_t)r * OUT_F;

    for (int cc = threadIdx.x; cc < OUT_F; cc += blockDim.x) {
        float acc = xr[cc];
#pragma unroll
        for (int i = 0; i < ORDER; ++i)
            acc -= c[i] * vn[(size_t)i * IN_F + cc];
        outr[cc] = acc * dvec[cc] + bvec[cc];
    }
}

// ---------------------------------------------------------------------------
extern "C" void kernel_launch(void* const* d_in, const int* in_sizes, int n_in,
                              void* d_out, int out_size, void* d_ws, size_t ws_size,
                              hipStream_t stream) {
    const float* x        = (const float*)d_in[0];
    const float* v_mean   = (const float*)d_in[1];
    const float* v_logvar = (const float*)d_in[2];
    const float* d_mean   = (const float*)d_in[3];
    const float* d_logvar = (const float*)d_in[4];
    const float* b_mean   = (const float*)d_in[5];
    const float* b_logvar = (const float*)d_in[6];
    const float* eps_v    = (const float*)d_in[7];
    const float* eps_d    = (const float*)d_in[8];
    const float* eps_b    = (const float*)d_in[9];
    float* out = (float*)d_out;

    // Workspace layout (floats):
    float* ws   = (float*)d_ws;
    float* vn   = ws;                         // 16 * IN_F   (rows 8..15 zero-pad)
    float* dvec = vn + (size_t)16 * IN_F;     // IN_F
    float* bvec = dvec + IN_F;                // OUT_F
    float* M8   = bvec + OUT_F;               // 64
    float* P    = M8 + 64;                    // N_ROWS * 8

    obd_setup<<<1, 256, 0, stream>>>(v_mean, v_logvar, d_mean, d_logvar,
                                     b_mean, b_logvar, eps_v, eps_d, eps_b,
                                     vn, dvec, bvec, M8);

    // 512 16-row tiles, WAVES tiles per block
    obd_pass1<<<N_ROWS / 16 / WAVES, 32 * WAVES, 0, stream>>>(x, vn, P);

    obd_pass2<<<N_ROWS, 256, 0, stream>>>(x, vn, P, M8, dvec, bvec, out);
}